// SimpleAttention_31009663877207
// MI455X (gfx1250) — compile-verified
//
#include <hip/hip_runtime.h>
#include <math.h>

typedef __attribute__((ext_vector_type(2))) float v2f;
typedef __attribute__((ext_vector_type(4))) float v4f;
typedef __attribute__((ext_vector_type(8))) float v8f;

#define NPIX 65536      // B*H*W = 16*64*64
#define CCH 256
#define PIXB 4096       // pixels per batch (H*W)

// ---------------------------------------------------------------------------
// Kernel 1: Mt[n][c] = sum_d Wk[n,d] * Wq[c,d]   (Mt = (Wq Wk^T)^T, 256x256)
// Stored transposed so the main kernel's B-fragments are contiguous in LDS.
// ---------------------------------------------------------------------------
__global__ __launch_bounds__(256) void prep_mt(const float* __restrict__ Wq,
                                               const float* __restrict__ Wk,
                                               float* __restrict__ Mt) {
    __shared__ float wkrow[CCH];
    const int n = blockIdx.x;
    const int c = threadIdx.x;
    wkrow[c] = Wk[n * CCH + c];
    __syncthreads();
    const float* wq = Wq + (size_t)c * CCH;
    float acc = 0.f;
#pragma unroll 8
    for (int d = 0; d < CCH; ++d) acc += wkrow[d] * wq[d];
    Mt[n * CCH + c] = acc;
}

// ---------------------------------------------------------------------------
// Kernel 1b: u[c] = sum_d (Wq[c,d]*bk[d] + Wk[c,d]*bq[d]);  wv[c] = sum_d Wv[c,d]
//            scalars[0] = bq.bk ; scalars[1] = sum(bv)
// ---------------------------------------------------------------------------
__global__ __launch_bounds__(256) void prep_vec(const float* __restrict__ Wq,
                                                const float* __restrict__ Wk,
                                                const float* __restrict__ Wv,
                                                const float* __restrict__ bq,
                                                const float* __restrict__ bk,
                                                const float* __restrict__ bv,
                                                float* __restrict__ u,
                                                float* __restrict__ wv,
                                                float* __restrict__ scalars) {
    __shared__ float red[256];
    const int c = threadIdx.x;
    float uu = 0.f, ww = 0.f;
    for (int d = 0; d < CCH; ++d) {
        uu += Wq[(size_t)c * CCH + d] * bk[d] + Wk[(size_t)c * CCH + d] * bq[d];
        ww += Wv[(size_t)c * CCH + d];
    }
    u[c] = uu;
    wv[c] = ww;

    red[c] = bq[c] * bk[c];
    __syncthreads();
    for (int s = 128; s > 0; s >>= 1) {
        if (c < s) red[c] += red[c + s];
        __syncthreads();
    }
    if (c == 0) scalars[0] = red[0];
    __syncthreads();
    red[c] = bv[c];
    __syncthreads();
    for (int s = 128; s > 0; s >>= 1) {
        if (c < s) red[c] += red[c + s];
        __syncthreads();
    }
    if (c == 0) scalars[1] = red[0];
}

// ---------------------------------------------------------------------------
// Kernel 2 (main, WMMA): per pixel p
//   score[p] = x[p]^T M x[p] + x[p].u + bqk      (M = Wq Wk^T)
//   sv[p]    = x[p].wv_sum + sum(bv)
// Y = X*M via v_wmma_f32_16x16x4_f32, fused row-dot against X.
// Block: 256 thr (8 waves), 128 rows.
// Mt staged in two 128KB halves -> 2 blocks/WGP possible.
// Rolled n-tile loop (<256 VGPRs); four independent WMMA accumulator chains;
// software-pipelined B-fragment loads (one DS group in flight over the WMMAs).
// ---------------------------------------------------------------------------
__global__ __launch_bounds__(256) void score_gemm(const float* __restrict__ X,
                                                  const float* __restrict__ Mt,
                                                  const float* __restrict__ u,
                                                  const float* __restrict__ wvsum,
                                                  const float* __restrict__ scalars,
                                                  float* __restrict__ score,
                                                  float* __restrict__ sv) {
    extern __shared__ float lds[];  // 32768 floats = half of Mt (128 rows)

    const int tid = threadIdx.x;
    const int lane = tid & 31;
    const int wave = tid >> 5;
    const int lo = lane & 15;   // A: row M / B,C,D: column N
    const int hi = lane >> 4;   // A: K-half  / C,D: row half
    const int row0 = blockIdx.x * 128 + wave * 16;

    // --- A fragments: 16 rows x 256 cols of X, ISA 16x4 fp32 layout ---
    // lane<16 holds K = 4k+0,4k+1 ; lane>=16 holds K = 4k+2,4k+3 (row = lo)
    v2f afrag[64];
    {
        const float* xr = X + (size_t)(row0 + lo) * CCH + hi * 2;
#pragma unroll
        for (int ks = 0; ks < 64; ++ks) afrag[ks] = *(const v2f*)(xr + ks * 4);
    }

    float part[8], xupart[8], wvpart[8];
#pragma unroll
    for (int r = 0; r < 8; ++r) { part[r] = 0.f; xupart[r] = 0.f; wvpart[r] = 0.f; }

    const v4f* src = (const v4f*)Mt;
    v4f* dst = (v4f*)lds;

#pragma unroll 1
    for (int half = 0; half < 2; ++half) {
        if (half) __syncthreads();  // all waves done reading previous half
        // stage 128 rows of Mt (coalesced b128 -> ds b128)
#pragma unroll
        for (int i = 0; i < 32; ++i)
            dst[i * 256 + tid] = src[(half * 32 + i) * 256 + tid];
        __syncthreads();

#pragma unroll 1
        for (int nth = 0; nth < 8; ++nth) {
            const int nt = half * 8 + nth;
            const int n = nt * 16 + lo;  // global output column
            // B fragment: b[r] = M[ks*4 + r + 2*hi][n] = Mt[n][ks*4 + r + 2*hi]
            const float* bbase = lds + (nth * 16 + lo) * CCH + hi * 2;

            v8f acc0 = {0.f, 0.f, 0.f, 0.f, 0.f, 0.f, 0.f, 0.f};
            v8f acc1 = {0.f, 0.f, 0.f, 0.f, 0.f, 0.f, 0.f, 0.f};
            v8f acc2 = {0.f, 0.f, 0.f, 0.f, 0.f, 0.f, 0.f, 0.f};
            v8f acc3 = {0.f, 0.f, 0.f, 0.f, 0.f, 0.f, 0.f, 0.f};

            // software pipeline: prefetch next K-group's B frags over the WMMAs
            v2f b0 = *(const v2f*)(bbase + 0);
            v2f b1 = *(const v2f*)(bbase + 4);
            v2f b2 = *(const v2f*)(bbase + 8);
            v2f b3 = *(const v2f*)(bbase + 12);
#pragma unroll
            for (int ks = 0; ks < 64; ks += 4) {
                const int nk = (ks + 4) & 63;  // last iter wraps (in-bounds reload)
                v2f nb0 = *(const v2f*)(bbase + nk * 4);
                v2f nb1 = *(const v2f*)(bbase + nk * 4 + 4);
                v2f nb2 = *(const v2f*)(bbase + nk * 4 + 8);
                v2f nb3 = *(const v2f*)(bbase + nk * 4 + 12);
                acc0 = __builtin_amdgcn_wmma_f32_16x16x4_f32(
                    false, afrag[ks], false, b0, (short)0, acc0, false, false);
                acc1 = __builtin_amdgcn_wmma_f32_16x16x4_f32(
                    false, afrag[ks + 1], false, b1, (short)0, acc1, false, false);
                acc2 = __builtin_amdgcn_wmma_f32_16x16x4_f32(
                    false, afrag[ks + 2], false, b2, (short)0, acc2, false, false);
                acc3 = __builtin_amdgcn_wmma_f32_16x16x4_f32(
                    false, afrag[ks + 3], false, b3, (short)0, acc3, false, false);
                b0 = nb0; b1 = nb1; b2 = nb2; b3 = nb3;
            }
            // Fuse: score partial = sum_n X[m][n]*Y[m][n]; plus x.u and x.wv dots.
            const float un = u[n];
            const float wn = wvsum[n];
#pragma unroll
            for (int r = 0; r < 8; ++r) {
                const int m = r + hi * 8;  // C/D layout: VGPR r = row r (+8 hi half)
                const float xv = X[(size_t)(row0 + m) * CCH + n];
                part[r]   += xv * ((acc0[r] + acc1[r]) + (acc2[r] + acc3[r]));
                xupart[r] += xv * un;
                wvpart[r] += xv * wn;
            }
        }
    }

    // reduce across the 16 lanes sharing each row group (xor<16 keeps halves apart)
#pragma unroll
    for (int r = 0; r < 8; ++r) {
#pragma unroll
        for (int off = 8; off > 0; off >>= 1) {
            part[r]   += __shfl_xor(part[r], off, 32);
            xupart[r] += __shfl_xor(xupart[r], off, 32);
            wvpart[r] += __shfl_xor(wvpart[r], off, 32);
        }
    }
    if (lo == 0) {
        const float bqk = scalars[0], bvs = scalars[1];
#pragma unroll
        for (int r = 0; r < 8; ++r) {
            const int m = r + hi * 8;
            score[row0 + m] = part[r] + xupart[r] + bqk;
            sv[row0 + m]    = wvpart[r] + bvs;
        }
    }
}

// ---------------------------------------------------------------------------
// Kernel 3: per-batch softmax stats over 4096 pixels -> {max, sumexp}
// ---------------------------------------------------------------------------
__global__ __launch_bounds__(256) void softmax_stats(const float* __restrict__ score,
                                                     float* __restrict__ stats) {
    __shared__ float red[256];
    const int b = blockIdx.x, t = threadIdx.x;
    const float* s = score + (size_t)b * PIXB;
    float m = -3.402823466e38f;
    for (int i = t; i < PIXB; i += 256) m = fmaxf(m, s[i]);
    red[t] = m;
    __syncthreads();
    for (int k = 128; k > 0; k >>= 1) {
        if (t < k) red[t] = fmaxf(red[t], red[t + k]);
        __syncthreads();
    }
    const float mb = red[0];
    __syncthreads();
    float sum = 0.f;
    for (int i = t; i < PIXB; i += 256) sum += expf(s[i] - mb);
    red[t] = sum;
    __syncthreads();
    for (int k = 128; k > 0; k >>= 1) {
        if (t < k) red[t] += red[t + k];
        __syncthreads();
    }
    if (t == 0) { stats[2 * b] = mb; stats[2 * b + 1] = red[0]; }
}

// ---------------------------------------------------------------------------
// Kernel 4: out[p,d] = attn[p]*sv[p]*Wo[d] + bo[d]   (64MB coalesced store)
// ---------------------------------------------------------------------------
__global__ __launch_bounds__(256) void final_out(const float* __restrict__ score,
                                                 const float* __restrict__ sv,
                                                 const float* __restrict__ stats,
                                                 const float* __restrict__ Wo,
                                                 const float* __restrict__ bo,
                                                 float* __restrict__ out) {
    const int idx = blockIdx.x * 256 + threadIdx.x;  // float4 index
    const int p = idx >> 6;               // 64 float4 per pixel row
    const int d4 = (idx & 63) * 4;
    const int b = p >> 12;                // 4096 pixels per batch
    const float a = expf(score[p] - stats[2 * b]) / stats[2 * b + 1] * sv[p];
    const v4f w = *(const v4f*)(Wo + d4);
    const v4f bb = *(const v4f*)(bo + d4);
    v4f o = a * w + bb;
    *(v4f*)(out + (size_t)p * CCH + d4) = o;
}

// ---------------------------------------------------------------------------
extern "C" void kernel_launch(void* const* d_in, const int* in_sizes, int n_in,
                              void* d_out, int out_size, void* d_ws, size_t ws_size,
                              hipStream_t stream) {
    const float* x  = (const float*)d_in[0];
    const float* Wq = (const float*)d_in[1];
    const float* bq = (const float*)d_in[2];
    const float* Wk = (const float*)d_in[3];
    const float* bk = (const float*)d_in[4];
    const float* Wv = (const float*)d_in[5];
    const float* bv = (const float*)d_in[6];
    const float* Wo = (const float*)d_in[7];
    const float* bo = (const float*)d_in[8];
    float* out = (float*)d_out;

    char* ws = (char*)d_ws;
    float* Mt      = (float*)(ws);                       // 256 KB
    float* score   = (float*)(ws + 262144);              // 256 KB
    float* sv      = (float*)(ws + 2 * 262144);          // 256 KB
    float* u       = (float*)(ws + 3 * 262144);          // 1 KB
    float* wv      = (float*)(ws + 3 * 262144 + 1024);   // 1 KB
    float* scalars = (float*)(ws + 3 * 262144 + 2048);   // 8 B
    float* stats   = (float*)(ws + 3 * 262144 + 2048 + 64); // 128 B

    prep_mt<<<256, 256, 0, stream>>>(Wq, Wk, Mt);
    prep_vec<<<1, 256, 0, stream>>>(Wq, Wk, Wv, bq, bk, bv, u, wv, scalars);
    score_gemm<<<NPIX / 128, 256, 131072, stream>>>(x, Mt, u, wv, scalars, score, sv);
    softmax_stats<<<16, 256, 0, stream>>>(score, stats);
    final_out<<<NPIX * 64 / 256, 256, 0, stream>>>(score, sv, stats, Wo, bo, out);
}